// ASA_38611755991794
// MI455X (gfx1250) — compile-verified
//
#include <hip/hip_runtime.h>
#include <math.h>

typedef __attribute__((ext_vector_type(2))) float v2f;
typedef __attribute__((ext_vector_type(8))) float v8f;

#define NN 50000
#define NE 640000
#define CH 128
#define CO 64

// ---------------- small utility kernels ----------------

__global__ void __launch_bounds__(256) k_fill_zero_i(int* __restrict__ p, int n) {
    int i = blockIdx.x * blockDim.x + threadIdx.x;
    if (i < n) p[i] = 0;
}

// in-degree histogram over dst (int atomics; reused across all 3 layers)
__global__ void __launch_bounds__(256) k_count(const int* __restrict__ dst,
                                               int* __restrict__ cnt, int e) {
    int i = blockIdx.x * blockDim.x + threadIdx.x;
    if (i < e)
        __hip_atomic_fetch_add(&cnt[dst[i]], 1, __ATOMIC_RELAXED,
                               __HIP_MEMORY_SCOPE_AGENT);
}

// dinv = rsqrt(deg+1), invd = 1/(deg+1)  from integer counts
__global__ void __launch_bounds__(256) k_deg_fin(const int* __restrict__ cnt,
                                                 float* __restrict__ dinv,
                                                 float* __restrict__ invd, int n) {
    int i = blockIdx.x * blockDim.x + threadIdx.x;
    if (i < n) {
        float d = (float)cnt[i] + 1.0f;
        invd[i] = 1.0f / d;
        dinv[i] = rsqrtf(d);
    }
}

// single-workgroup exclusive scan: cnt[0..n) -> rowstart[0..n]  (rowstart[n] = total)
__global__ void __launch_bounds__(1024) k_scan(const int* __restrict__ cnt,
                                               int* __restrict__ rowstart, int n) {
    __shared__ int buf[1024];
    __shared__ int carry;
    const int tid = threadIdx.x;
    if (tid == 0) carry = 0;
    __syncthreads();
    for (int base = 0; base < n; base += 1024) {
        int i = base + tid;
        int v = (i < n) ? cnt[i] : 0;
        buf[tid] = v;
        __syncthreads();
#pragma unroll
        for (int off = 1; off < 1024; off <<= 1) {   // Hillis-Steele inclusive scan
            int t = (tid >= off) ? buf[tid - off] : 0;
            __syncthreads();
            buf[tid] += t;
            __syncthreads();
        }
        int c = carry;                                // everyone reads old carry
        if (i < n) rowstart[i] = c + buf[tid] - v;    // exclusive
        __syncthreads();
        if (tid == 0) carry = c + buf[1023];
        __syncthreads();
    }
    if (tid == 0) rowstart[n] = carry;
}

// permute edges into CSR-by-dst; store src index and normalized coefficient
__global__ void __launch_bounds__(256) k_csr_fill(const int* __restrict__ src,
                                                  const int* __restrict__ dst,
                                                  const float* __restrict__ dinv,
                                                  const int* __restrict__ rowstart,
                                                  int* __restrict__ cursor,
                                                  int* __restrict__ csr_src,
                                                  float* __restrict__ csr_coef, int e) {
    int i = blockIdx.x * blockDim.x + threadIdx.x;
    if (i >= e) return;
    int s = src[i], d = dst[i];
    int pos = rowstart[d] + __hip_atomic_fetch_add(&cursor[d], 1, __ATOMIC_RELAXED,
                                                   __HIP_MEMORY_SCOPE_AGENT);
    csr_src[pos]  = s;
    csr_coef[pos] = dinv[s] * dinv[d];
}

// ---------------- fp32 WMMA GEMM: H[rows x 16*NT] = act(X[rows x K]) @ W ----------
// One wave computes a full 16-row x (16*NT)-col strip: A fragment loaded once per
// K-step, feeds NT independent WMMAs (no D->A/B hazard, back-to-back issue).

template <int K, int NT, bool RELU_IN>
__global__ void __launch_bounds__(256) k_gemm_wmma(const float* __restrict__ X,
                                                   const float* __restrict__ W,
                                                   float* __restrict__ H, int rows) {
    constexpr int COUT = NT * 16;
    const int lane  = threadIdx.x & 31;
    const int wave  = threadIdx.x >> 5;
    const int mTile = blockIdx.x * (blockDim.x >> 5) + wave;
    if (mTile >= (rows >> 4)) return;      // whole-wave exit; EXEC all-ones otherwise

    const int m15   = lane & 15;           // M (for A) / N (for B,C,D)
    const int khalf = (lane >> 4) * 2;     // K offset 0 or 2 per ISA A/B layout

    const float* __restrict__ xrow  = X + (mTile * 16 + m15) * K;
    const float* __restrict__ wbase = W + m15;

    v8f acc[NT] = {};
#pragma unroll 2
    for (int k = 0; k < K; k += 4) {
        float a0 = xrow[k + khalf];
        float a1 = xrow[k + khalf + 1];
        if (RELU_IN) { a0 = fmaxf(a0, 0.0f); a1 = fmaxf(a1, 0.0f); }
        v2f a; a.x = a0; a.y = a1;
        const float* w0 = wbase + (k + khalf) * COUT;
        const float* w1 = wbase + (k + khalf + 1) * COUT;
#pragma unroll
        for (int t = 0; t < NT; ++t) {
            v2f b; b.x = w0[t * 16]; b.y = w1[t * 16];
            // 8 args: (neg_a, A, neg_b, B, c_mod, C, reuse_a, reuse_b)
            acc[t] = __builtin_amdgcn_wmma_f32_16x16x4_f32(false, a, false, b,
                                                           (short)0, acc[t],
                                                           false, false);
        }
    }
    // C/D layout: VGPR r -> row mTile*16 + r (+8 for lanes 16..31), col t*16 + m15
    const int rowBase = mTile * 16 + (lane >> 4) * 8;
#pragma unroll
    for (int t = 0; t < NT; ++t) {
        float* __restrict__ hp = H + rowBase * COUT + t * 16 + m15;
#pragma unroll
        for (int r = 0; r < 8; ++r) hp[r * COUT] = acc[t][r];
    }
}

// ---------------- CSR pull-aggregation (no atomics) ----------------
// One wave per node: acc = h[node]/deg + b, then acc += coef_j * h[src_j] over the
// node's incoming edges; single coalesced row write. h (25.6MB) is L2-resident.

template <int C>
__global__ void __launch_bounds__(256) k_aggregate(const float* __restrict__ h,
                                                   const int* __restrict__ rowstart,
                                                   const int* __restrict__ csr_src,
                                                   const float* __restrict__ csr_coef,
                                                   const float* __restrict__ invd,
                                                   const float* __restrict__ bias,
                                                   float* __restrict__ out, int n) {
    constexpr int V = C / 32;   // floats per lane: 4 (C=128) or 2 (C=64)
    const int lane = threadIdx.x & 31;
    const int node = (blockIdx.x * blockDim.x + threadIdx.x) >> 5;
    if (node >= n) return;
    const int ch = lane * V;

    float acc[V];
    {
        const float iv = invd[node];
        if constexpr (V == 4) {
            float4 hv = *reinterpret_cast<const float4*>(h + node * C + ch);
            float4 bv = *reinterpret_cast<const float4*>(bias + ch);
            acc[0] = hv.x * iv + bv.x; acc[1] = hv.y * iv + bv.y;
            acc[2] = hv.z * iv + bv.z; acc[3] = hv.w * iv + bv.w;
        } else {
            float2 hv = *reinterpret_cast<const float2*>(h + node * C + ch);
            float2 bv = *reinterpret_cast<const float2*>(bias + ch);
            acc[0] = hv.x * iv + bv.x; acc[1] = hv.y * iv + bv.y;
        }
    }
    const int e0 = rowstart[node], e1 = rowstart[node + 1];
    for (int j = e0; j < e1; ++j) {
        const int   s  = csr_src[j];
        const float cf = csr_coef[j];
        if constexpr (V == 4) {
            float4 hv = *reinterpret_cast<const float4*>(h + s * C + ch);
            acc[0] += cf * hv.x; acc[1] += cf * hv.y;
            acc[2] += cf * hv.z; acc[3] += cf * hv.w;
        } else {
            float2 hv = *reinterpret_cast<const float2*>(h + s * C + ch);
            acc[0] += cf * hv.x; acc[1] += cf * hv.y;
        }
    }
    if constexpr (V == 4) {
        float4 o; o.x = acc[0]; o.y = acc[1]; o.z = acc[2]; o.w = acc[3];
        *reinterpret_cast<float4*>(out + node * C + ch) = o;
    } else {
        float2 o; o.x = acc[0]; o.y = acc[1];
        *reinterpret_cast<float2*>(out + node * C + ch) = o;
    }
}

// ---------------- host-side orchestration ----------------

static inline int cdiv(int a, int b) { return (a + b - 1) / b; }

extern "C" void kernel_launch(void* const* d_in, const int* in_sizes, int n_in,
                              void* d_out, int out_size, void* d_ws, size_t ws_size,
                              hipStream_t stream) {
    const float* x  = (const float*)d_in[0];
    const int*   ei = (const int*)d_in[1];
    const float* W1 = (const float*)d_in[2];
    const float* b1 = (const float*)d_in[3];
    const float* W2 = (const float*)d_in[4];
    const float* b2 = (const float*)d_in[5];
    const float* W3 = (const float*)d_in[6];
    const float* b3 = (const float*)d_in[7];
    float* out = (float*)d_out;

    const int* src = ei;            // edge_index[0]
    const int* dst = ei + NE;       // edge_index[1]

    // workspace layout (floats/ints are both 4 bytes)
    float* A        = (float*)d_ws;              // N x 128  (GEMM result h)
    float* B        = A + NN * CH;               // N x 128  (layer output)
    float* dinv     = B + NN * CH;               // N  rsqrt(deg)
    float* invd     = dinv + NN;                 // N  1/deg
    float* csr_coef = invd + NN;                 // E
    int*   csr_src  = (int*)(csr_coef + NE);     // E
    int*   cnt      = csr_src + NE;              // N   \ zeroed together
    int*   cursor   = cnt + NN;                  // N   /
    int*   rowstart = cursor + NN;               // N + 1

    const int T = 256;

    // ---- CSR build + normalization coefficients (edge structure is layer-invariant)
    k_fill_zero_i<<<cdiv(2 * NN, T), T, 0, stream>>>(cnt, 2 * NN);  // cnt + cursor
    k_count<<<cdiv(NE, T), T, 0, stream>>>(dst, cnt, NE);
    k_scan<<<1, 1024, 0, stream>>>(cnt, rowstart, NN);
    k_deg_fin<<<cdiv(NN, T), T, 0, stream>>>(cnt, dinv, invd, NN);
    k_csr_fill<<<cdiv(NE, T), T, 0, stream>>>(src, dst, dinv, rowstart, cursor,
                                              csr_src, csr_coef, NE);

    const int wavesPerBlk = T / 32;
    const int mTiles   = NN / 16;                      // 3125 (exact)
    const int gemmBlks = cdiv(mTiles, wavesPerBlk);
    const int aggBlks128 = cdiv(NN * 32, T);           // one wave per node
    const int aggBlks64  = cdiv(NN * 32, T);

    // ---- layer 1: h = x @ W1 ; B = agg(h) + h/deg + b1  (relu fused into next gemm)
    k_gemm_wmma<CH, CH / 16, false><<<gemmBlks, T, 0, stream>>>(x, W1, A, NN);
    k_aggregate<CH><<<aggBlks128, T, 0, stream>>>(A, rowstart, csr_src, csr_coef,
                                                  invd, b1, B, NN);
    // ---- layer 2: h = relu(B) @ W2
    k_gemm_wmma<CH, CH / 16, true><<<gemmBlks, T, 0, stream>>>(B, W2, A, NN);
    k_aggregate<CH><<<aggBlks128, T, 0, stream>>>(A, rowstart, csr_src, csr_coef,
                                                  invd, b2, B, NN);
    // ---- layer 3: h = relu(B) @ W3 ; final output (no relu)
    k_gemm_wmma<CH, CO / 16, true><<<gemmBlks, T, 0, stream>>>(B, W3, A, NN);
    k_aggregate<CO><<<aggBlks64, T, 0, stream>>>(A, rowstart, csr_src, csr_coef,
                                                 invd, b3, out, NN);
}